// projection_layer_vm_learned_20091857011278
// MI455X (gfx1250) — compile-verified
//
#include <hip/hip_runtime.h>

#define ND    6
#define NVM   4
#define FDIM  128
#define NH    16
#define XROW  (ND * NVM * FDIM)   // 3072 floats = 12288 bytes per row

typedef float v2f __attribute__((ext_vector_type(2)));
typedef float v8f __attribute__((ext_vector_type(8)));
typedef unsigned int v4u __attribute__((ext_vector_type(4)));
typedef int v4i __attribute__((ext_vector_type(4)));
typedef int v8i __attribute__((ext_vector_type(8)));

#if __has_builtin(__builtin_amdgcn_tensor_load_to_lds)
#define HAVE_TDM 1
#else
#define HAVE_TDM 0
#endif
#if __has_builtin(__builtin_amdgcn_wmma_f32_16x16x4_f32)
#define HAVE_WMMA_F32X4 1
#else
#define HAVE_WMMA_F32X4 0
#endif

#if HAVE_TDM
// Issue a 1-D TDM transfer of one x-row (3072 fp32) into LDS offset `ldsByteOff`.
// D# bitfields per CDNA5 ISA ch. 8 (group0: flags/lds/global addr, group1: dims).
// amdgpu-toolchain (clang-23) builtin: 6 args
//   (uint32x4 g0, int32x8 g1, int32x4, int32x4, int32x8, i32 cpol)
__device__ __forceinline__ void tdm_load_x_row(const float* gsrc, unsigned ldsByteOff) {
  unsigned long long ga = (unsigned long long)(size_t)gsrc;
  v4u g0;
  g0.x = 1u;                                     // count=1, user descriptor
  g0.y = ldsByteOff;                             // lds_addr (bytes)
  g0.z = (unsigned)(ga & 0xFFFFFFFFu);           // global_addr[31:0]
  g0.w = (unsigned)((ga >> 32) & 0x1FFFFFFu)     // global_addr[56:32]
       | 0x80000000u;                            // type = 2 ("image") in bits 127:126
  v8i g1;
  const unsigned ne = XROW;                      // 3072 elements
  g1[0] = (int)(2u << 16);                       // workgroup_mask=0, data_size=2 (4B)
  g1[1] = (int)((ne & 0xFFFFu) << 16);           // tensor_dim0[15:0] @ bits 63:48
  g1[2] = (int)((ne >> 16) & 0xFFFFu) | (1 << 16); // tensor_dim0[31:16], tensor_dim1=1
  g1[3] = (int)((ne & 0xFFFFu) << 16);           // tile_dim0 @ bits 127:112
  g1[4] = 1;                                     // tile_dim1 = 1
  g1[5] = (int)ne;                               // tensor_dim0_stride[31:0]
  g1[6] = 0;
  g1[7] = 0;
  v4i z4 = {0, 0, 0, 0};
  v8i z8 = {0, 0, 0, 0, 0, 0, 0, 0};
  __builtin_amdgcn_tensor_load_to_lds(g0, g1, z4, z4, z8, 0);
}
#endif

__global__ void __launch_bounds__(128)
proj_vm_kernel(const float* __restrict__ x,
               const float* __restrict__ oc,
               const float* __restrict__ pamp,
               const float* __restrict__ pwl,
               const float* __restrict__ dwp,
               const float* __restrict__ phi0,
               const float* __restrict__ dists0,
               float* __restrict__ out,
               int nrows) {
  extern __shared__ float smem[];
  float* lds_x   = smem;            // 3072 floats (later reused as arg[16][128])
  float* lds_sm  = smem + 3072;     // 512  : softmax exchange for scrambled reshape
  float* lds_up  = smem + 3584;     // 512  : B operand, interleaved {u1,u2}, zero-padded
  float* lds_amp = smem + 4096;     // 128
  float* lds_off = smem + 4224;     // 128
  float* lds_vp  = smem + 4352;     // 64   : A operand, interleaved {v1,v2}, zero-padded
                                    // total 4416 floats = 17664 B

  const int row = blockIdx.x;
  if (row >= nrows) return;
  const int tid = threadIdx.x;
  const int fi  = tid;              // one feature per thread

  // ---------- stage x row into LDS via TDM ----------
#if HAVE_TDM
  if (tid < 32) {                   // wave 0 issues the DMA (TENSORcnt is per-wave)
    tdm_load_x_row(x + (size_t)row * XROW, 0u);
  }
#else
  {
    const float* gx = x + (size_t)row * XROW;
    for (int idx = tid; idx < XROW; idx += 128) lds_x[idx] = gx[idx];
  }
#endif

  // ---------- softmax(dist_weights_phi): uniform, overlaps the DMA ----------
  float wd[ND - 1];
  {
    float m = dwp[0];
#pragma unroll
    for (int j = 1; j < ND - 1; ++j) m = fmaxf(m, dwp[j]);
    float s = 0.f;
#pragma unroll
    for (int j = 0; j < ND - 1; ++j) { wd[j] = expf(dwp[j] - m); s += wd[j]; }
    float inv = 1.f / s;
#pragma unroll
    for (int j = 0; j < ND - 1; ++j) wd[j] *= inv;
  }

  // ---------- haversine distance/bearing per head (independent of x) ----------
  // A-operand of the rank-2 WMMA: lanes 0-15 = {dist*cos(phi), dist*sin(phi)},
  // lanes 16-31 = zero (K=2,3 padding) -> stored pre-padded, no exec-masked loads.
  if (tid < NH) {
    const size_t cs   = (size_t)nrows * NH;      // coordinate channel stride
    const size_t base = (size_t)row * NH;
    float lon1 = oc[base];
    float lat1 = oc[cs + base];
    float lon2 = oc[base + tid];
    float lat2 = oc[cs + base + tid];
    float dlon = lon2 - lon1, dlat = lat2 - lat1;
    float sdlat = sinf(0.5f * dlat), sdlon = sinf(0.5f * dlon);
    float a = sdlat * sdlat + cosf(lat1) * cosf(lat2) * sdlon * sdlon;
    a = fminf(fmaxf(a, 0.f), 1.f) + 1e-12f;
    float dist = 2.f * asinf(sqrtf(a));
    float py = sinf(dlon) * cosf(lat2);
    float px = cosf(lat1) * sinf(lat2) - sinf(lat1) * cosf(lat2) * cosf(dlon);
    float phi = atan2f(py, px);
    lds_vp[2 * tid + 0] = dist * cosf(phi);
    lds_vp[2 * tid + 1] = dist * sinf(phi);
  } else if (tid < 32) {            // zero K=2,3 lanes of A
    lds_vp[2 * tid + 0] = 0.f;
    lds_vp[2 * tid + 1] = 0.f;
  }

#if HAVE_TDM
  if (tid < 32) __builtin_amdgcn_s_wait_tensorcnt(0);
#endif
  __syncthreads();

  // ---------- per-row math, one feature per thread ----------
  float xv[ND][NVM];
#pragma unroll
  for (int d = 0; d < ND; ++d)
#pragma unroll
    for (int v = 0; v < NVM; ++v)
      xv[d][v] = lds_x[d * (NVM * FDIM) + v * FDIM + fi];

  float xoff = 0.25f * (xv[0][0] + xv[0][1] + xv[0][2] + xv[0][3]);

  float dp[NVM];
#pragma unroll
  for (int v = 0; v < NVM; ++v) {
    float s = 0.f;
#pragma unroll
    for (int d = 0; d < ND - 1; ++d) s += wd[d] * xv[d + 1][v];
    dp[v] = s;
  }
  float m4 = fmaxf(fmaxf(dp[0], dp[1]), fmaxf(dp[2], dp[3]));
  float e4[NVM], s4 = 0.f;
#pragma unroll
  for (int v = 0; v < NVM; ++v) { e4[v] = expf(dp[v] - m4); s4 += e4[v]; }
  float inv4 = 1.f / s4;
  float dsc = 0.f;
#pragma unroll
  for (int v = 0; v < NVM; ++v) {
    float sv = e4[v] * inv4;
    lds_sm[v * FDIM + fi] = sv;       // softmax over vm, kept per (vm, fi)
    dsc += sv * cosf(phi0[v]);        // phi_0[0][vm]
  }
  float dirc = acosf(fminf(fmaxf(dsc, -1.f), 1.f)) + 1.5707963267948966f;
  __syncthreads();

  // scrambled dir_w reshape: weight for (vm', fi) = sm[fi%4][vm'*32 + fi/4]
  float wq[NVM];
#pragma unroll
  for (int v = 0; v < NVM; ++v)
    wq[v] = lds_sm[(fi & 3) * FDIM + v * 32 + (fi >> 2)];

  float xw[ND];
#pragma unroll
  for (int d = 0; d < ND; ++d) {
    float s = 0.f;
#pragma unroll
    for (int v = 0; v < NVM; ++v) s += xv[d][v] * wq[v];
    xw[d] = s;
  }

  float m6 = xw[0];
#pragma unroll
  for (int d = 1; d < ND; ++d) m6 = fmaxf(m6, xw[d]);
  float e6[ND], s6 = 0.f;
#pragma unroll
  for (int d = 0; d < ND; ++d) { e6[d] = expf(xw[d] - m6); s6 += e6[d]; }
  float inv6 = 1.f / s6;
  float t = 0.f, amp = 0.f;
#pragma unroll
  for (int d = 0; d < ND; ++d) {
    t   += (e6[d] * inv6) * dists0[d * NVM] * pwl[d];
    amp += (xw[d] - xoff) * pamp[d];
  }
  float wlv = 1.f / (1.f + expf(-t)) + 0.01f;
  float K = 6.283185307179586f / wlv;

  // B operand, interleaved + pre-padded: tile (fi>>4), col (fi&15);
  // lanes 16-31 of each tile (K=2,3) zeroed explicitly.
  {
    int ub = ((fi >> 4) * 32 + (fi & 15)) * 2;
    lds_up[ub + 0]  = K * cosf(dirc);
    lds_up[ub + 1]  = K * sinf(dirc);
    lds_up[ub + 32] = 0.f;
    lds_up[ub + 33] = 0.f;
  }
  lds_amp[fi] = amp;
  lds_off[fi] = xoff;
  __syncthreads();

  // ---------- head expansion: arg[h][fi] = u1[fi]*v1[h] + u2[fi]*v2[h] ----------
  float* lds_arg = lds_x;             // x row no longer needed; reuse (16*128 floats)
#if HAVE_WMMA_F32X4
  {
    const int lane = tid & 31;
    const int wv   = tid >> 5;
    v2f av = *(const v2f*)&lds_vp[lane * 2];    // one ds_load_b64, full EXEC
    v8f cz = {0.f, 0.f, 0.f, 0.f, 0.f, 0.f, 0.f, 0.f};
#pragma unroll
    for (int tt = 0; tt < 2; ++tt) {  // 8 feature tiles of 16, 2 per wave
      int tile = wv * 2 + tt;
      v2f bv = *(const v2f*)&lds_up[(tile * 32 + lane) * 2];
      v8f dacc = __builtin_amdgcn_wmma_f32_16x16x4_f32(
          false, av, false, bv, (short)0, cz, false, false);
      int fcol  = tile * 16 + (lane & 15);
      int hbase = (lane >> 4) * 8;    // D layout: VGPR r -> M=r / M=r+8 per lane half
#pragma unroll
      for (int r = 0; r < 8; ++r)
        lds_arg[(hbase + r) * FDIM + fcol] = dacc[r];
    }
  }
#else
  {
#pragma unroll
    for (int h = 0; h < NH; ++h) {
      int ub = ((fi >> 4) * 32 + (fi & 15)) * 2;
      lds_arg[h * FDIM + fi] =
          lds_up[ub] * lds_vp[2 * h] + lds_up[ub + 1] * lds_vp[2 * h + 1];
    }
  }
#endif
  __syncthreads();

  // ---------- final: fully-coalesced stores, one feature per thread ----------
  {
    float ampv = lds_amp[fi], offv = lds_off[fi];
    size_t obase = (size_t)row * NH * FDIM;
#pragma unroll
    for (int h = 0; h < NH; ++h) {
      float arg = lds_arg[h * FDIM + fi];
      out[obase + (size_t)h * FDIM + fi] = ampv * cosf(arg) + offv;
    }
  }
}

extern "C" void kernel_launch(void* const* d_in, const int* in_sizes, int n_in,
                              void* d_out, int out_size, void* d_ws, size_t ws_size,
                              hipStream_t stream) {
  (void)n_in; (void)out_size; (void)d_ws; (void)ws_size;
  const float* x      = (const float*)d_in[0];
  const float* oc     = (const float*)d_in[1];
  const float* pamp   = (const float*)d_in[2];
  const float* pwl    = (const float*)d_in[3];
  const float* dwp    = (const float*)d_in[4];
  const float* phi0   = (const float*)d_in[5];
  const float* dists0 = (const float*)d_in[6];
  float* out = (float*)d_out;

  const int nrows = in_sizes[0] / XROW;       // b*n = 16384
  const size_t shmem = 4416 * sizeof(float);  // 17664 B

  proj_vm_kernel<<<nrows, 128, shmem, stream>>>(
      x, oc, pamp, pwl, dwp, phi0, dists0, out, nrows);
}